// Transducer_89902255439970
// MI455X (gfx1250) — compile-verified
//
#include <hip/hip_runtime.h>
#include <hip/hip_bf16.h>
#include <stdint.h>

#define T_FRAMES 2000
#define B_BATCH  64
#define C_VOCAB  256
#define L_TGT    200
#define NEG_INF_F (-1.0e30f)
#define LOG2E_F  1.44269504088896340736f
#define LN2_F    0.69314718055994530942f

typedef __attribute__((ext_vector_type(2))) float v2f;
typedef __attribute__((ext_vector_type(8))) float v8f;

#ifndef __has_builtin
#define __has_builtin(x) 0
#endif

// ---------------- async global -> LDS copy (CDNA5) ----------------
#if __has_builtin(__builtin_amdgcn_global_load_async_to_lds_b128)
#define HAVE_ASYNC_BUILTIN 1
#else
#define HAVE_ASYNC_BUILTIN 0
#endif

typedef int v4i_vs __attribute__((__vector_size__(4 * sizeof(int))));
typedef __attribute__((address_space(1))) v4i_vs* gas_v4p;   // global int4*
typedef __attribute__((address_space(3))) v4i_vs* las_v4p;   // LDS int4*

__device__ __forceinline__ void async_copy_b128(const float* gp, float* lp) {
#if HAVE_ASYNC_BUILTIN
  __builtin_amdgcn_global_load_async_to_lds_b128((gas_v4p)(uintptr_t)gp,
                                                 (las_v4p)(uintptr_t)lp, 0, 0);
#else
  unsigned lds_off = (unsigned)(uintptr_t)lp;         // low 32 bits == LDS offset
  unsigned long long ga = (unsigned long long)(uintptr_t)gp;
  asm volatile("global_load_async_to_lds_b128 %0, %1, off"
               :: "v"(lds_off), "v"(ga) : "memory");
#endif
}

__device__ __forceinline__ void wait_async0() {
#if __has_builtin(__builtin_amdgcn_s_wait_asynccnt)
  __builtin_amdgcn_s_wait_asynccnt(0);
#else
  asm volatile("s_wait_asynccnt 0" ::: "memory");
#endif
  asm volatile("" ::: "memory");
}

// Native base-2 transcendentals (v_exp_f32 / v_log_f32 are base-2 on AMDGPU).
__device__ __forceinline__ float exp2_fast(float x) {
#if __has_builtin(__builtin_amdgcn_exp2f)
  return __builtin_amdgcn_exp2f(x);
#else
  return exp2f(x);
#endif
}
__device__ __forceinline__ float log2_fast(float x) {
#if __has_builtin(__builtin_amdgcn_logf)
  return __builtin_amdgcn_logf(x);
#else
  return log2f(x);
#endif
}

// ---------------- Kernel 1: row-wise logsumexp over C=256 ----------------
// One wave handles 16 rows (matches WMMA 16x16x4 f32 A layout).
// Lane L: row = L%16, half = L/16 (columns half*128 .. half*128+127).
#define K1_WAVES 2
#define ROWS_PER_WAVE 16
#define ROW_LDS_F 260   // 256 floats + 4-dword pad (bank-conflict avoidance)

__global__ __launch_bounds__(64) void lse_kernel(const float* __restrict__ inp,
                                                 float* __restrict__ lse) {
  __shared__ __align__(16) float smem[K1_WAVES * ROWS_PER_WAVE * ROW_LDS_F];
  const int lane = threadIdx.x & 31;
  const int w    = threadIdx.x >> 5;
  const int rowBase = (blockIdx.x * K1_WAVES + w) * ROWS_PER_WAVE;
  float* wls = smem + w * (ROWS_PER_WAVE * ROW_LDS_F);
  const float* gsrc = inp + (size_t)rowBase * C_VOCAB;

  // Stage 16 rows x 1KB into LDS, coalesced 512B per async b128 instruction.
  #pragma unroll
  for (int i = 0; i < 32; ++i) {
    int chunk = i * 32 + lane;      // 16-byte chunks, 64 per row
    int r = chunk >> 6;
    int c = chunk & 63;
    async_copy_b128(gsrc + chunk * 4, wls + r * ROW_LDS_F + c * 4);
  }
  wait_async0();

  const int r = lane & 15;
  const int h = lane >> 4;
  const float4* src4 = (const float4*)(wls + r * ROW_LDS_F + h * 128);

  // Pass A: per-half row max, then combine halves across lane pair (L, L+16).
  float m = NEG_INF_F;
  #pragma unroll
  for (int j = 0; j < 32; ++j) {
    float4 v = src4[j];
    m = fmaxf(m, fmaxf(fmaxf(v.x, v.y), fmaxf(v.z, v.w)));
  }
  m = fmaxf(m, __shfl_xor(m, 16, 32));

  // Pass B: two running partial sums of 2^((x-m)*log2e) per lane.
  float s0 = 0.f, s1 = 0.f;
  #pragma unroll
  for (int j = 0; j < 32; ++j) {
    float4 v = src4[j];
    s0 += exp2_fast((v.x - m) * LOG2E_F) + exp2_fast((v.y - m) * LOG2E_F);
    s1 += exp2_fast((v.z - m) * LOG2E_F) + exp2_fast((v.w - m) * LOG2E_F);
  }

#if __has_builtin(__builtin_amdgcn_wmma_f32_16x16x4_f32)
  // A(16x4): lane L%16 row, (s0,s1) in the two A VGPRs; B = ones(4x16).
  // D[m][n] = s0(lane m)+s1(lane m)+s0(lane m+16)+s1(lane m+16) = row-m total.
  v2f a; a.x = s0; a.y = s1;
  v2f bb; bb.x = 1.0f; bb.y = 1.0f;
  v8f c = {};
  v8f d = __builtin_amdgcn_wmma_f32_16x16x4_f32(false, a, false, bb,
                                                (short)0, c, false, false);
  // D layout: rows 0-7 -> VGPR m, lanes 0-15; rows 8-15 -> VGPR m-8, lanes 16-31.
  int idx = -1, orow = 0;
  if (lane < 8)        { idx = lane;      orow = lane;      }  // has row `lane` max
  else if (lane >= 24) { idx = lane - 24; orow = lane - 16; }  // has row lane-16 max
  if (idx >= 0) {
    float sum = d[0];
    #pragma unroll
    for (int q = 1; q < 8; ++q) sum = (idx == q) ? d[q] : sum;
    lse[rowBase + orow] = m + log2_fast(sum) * LN2_F;   // back to natural log
  }
#else
  float s = s0 + s1;
  s += __shfl_xor(s, 16, 32);
  if (lane < 16) lse[rowBase + lane] = m + log2_fast(s) * LN2_F;
#endif
}

// ---------------- Kernel 2: sequential forward DP, one wave per batch ----------------
// Entire DP runs in the log2 domain: v_exp_f32/v_log_f32 are natively base-2,
// so lae2 is max,min,sub,exp,add,log,add with no range fixups (arg of log in
// (1,2], arg of exp <= 0). Emissions are pre-scaled by log2(e) at prefetch.
__device__ __forceinline__ float lae2(float p, float q) {   // log2(2^p + 2^q)
  float m = fmaxf(p, q);
  float d = fminf(p, q) - m;                 // <= 0
  return m + log2_fast(1.0f + exp2_fast(d));
}

#define DP_V 7     // tokens per lane (32*7 >= 200)
#define DP_CH 8    // software-pipeline depth (frames prefetched)

__global__ __launch_bounds__(32) void dp_kernel(const float* __restrict__ inp,
                                                const float* __restrict__ lse,
                                                const int* __restrict__ tgt,
                                                float* __restrict__ alphaOut) {
  const int b = blockIdx.x;
  const int lane = threadIdx.x;

  int tg[DP_V];
  #pragma unroll
  for (int j = 0; j < DP_V; ++j) {
    int l = lane * DP_V + j;
    tg[j] = (l < L_TGT) ? tgt[b * L_TGT + l] : 0;
  }

  float a[DP_V];
  #pragma unroll
  for (int j = 0; j < DP_V; ++j) a[j] = NEG_INF_F;
  if (lane == 0)
    a[0] = (inp[(size_t)b * C_VOCAB + tg[0]] - lse[b]) * LOG2E_F;   // t = 0

  float xb[DP_CH][DP_V];
  // Prologue: prefetch emissions for t = 1..DP_CH (log2-scaled).
  #pragma unroll
  for (int u = 0; u < DP_CH; ++u) {
    int t = 1 + u;
    const float* rowp = inp + (size_t)(t * B_BATCH + b) * C_VOCAB;
    float ls = lse[t * B_BATCH + b];          // block-uniform -> s_load
    #pragma unroll
    for (int j = 0; j < DP_V; ++j) xb[u][j] = (rowp[tg[j]] - ls) * LOG2E_F;
  }

  for (int tc = 1; tc < T_FRAMES; tc += DP_CH) {
    float xc[DP_CH][DP_V];
    #pragma unroll
    for (int u = 0; u < DP_CH; ++u)
      #pragma unroll
      for (int j = 0; j < DP_V; ++j) xc[u][j] = xb[u][j];

    int tn = tc + DP_CH;
    if (tn < T_FRAMES) {                       // prefetch next chunk
      #pragma unroll
      for (int u = 0; u < DP_CH; ++u) {
        int t = tn + u;
        if (t < T_FRAMES) {
          const float* rowp = inp + (size_t)(t * B_BATCH + b) * C_VOCAB;
          float ls = lse[t * B_BATCH + b];
          #pragma unroll
          for (int j = 0; j < DP_V; ++j) xb[u][j] = (rowp[tg[j]] - ls) * LOG2E_F;
        }
      }
    }

    #pragma unroll
    for (int u = 0; u < DP_CH; ++u) {
      if (tc + u >= T_FRAMES) break;
      float pb = __shfl_up(a[DP_V - 1], 1, 32);   // neighbor lane's last token
      if (lane == 0) pb = NEG_INF_F;
      float nj[DP_V];
      nj[0] = xc[u][0] + lae2(a[0], pb);
      #pragma unroll
      for (int j = 1; j < DP_V; ++j) nj[j] = xc[u][j] + lae2(a[j], a[j - 1]);
      #pragma unroll
      for (int j = 0; j < DP_V; ++j) a[j] = nj[j];
    }
  }

  if (lane == (L_TGT - 1) / DP_V)                 // lane 28 holds l = 199
    alphaOut[b] = a[(L_TGT - 1) % DP_V] * LN2_F;  // back to natural log
}

// ---------------- Kernel 3: loss = mean_b(-alpha[b, L-1]) ----------------
__global__ void finalize_kernel(const float* __restrict__ alphaB,
                                float* __restrict__ out) {
  if (threadIdx.x == 0) {
    float s = 0.f;
    for (int i = 0; i < B_BATCH; ++i) s += alphaB[i];
    out[0] = -s * (1.0f / (float)B_BATCH);
  }
}

// ---------------- launch ----------------
extern "C" void kernel_launch(void* const* d_in, const int* in_sizes, int n_in,
                              void* d_out, int out_size, void* d_ws, size_t ws_size,
                              hipStream_t stream) {
  const float* inp = (const float*)d_in[0];   // [T, B, C] f32
  const int*   tgt = (const int*)d_in[1];     // [B, L] int
  float* lse    = (float*)d_ws;                         // T*B floats
  float* alphaB = lse + (size_t)T_FRAMES * B_BATCH;     // B floats

  const int rows = T_FRAMES * B_BATCH;                  // 128000
  lse_kernel<<<rows / (K1_WAVES * ROWS_PER_WAVE), K1_WAVES * 32, 0, stream>>>(inp, lse);
  dp_kernel<<<B_BATCH, 32, 0, stream>>>(inp, lse, tgt, alphaB);
  finalize_kernel<<<1, 32, 0, stream>>>(alphaB, (float*)d_out);
}